// GCN_29772713296319
// MI455X (gfx1250) — compile-verified
//
#include <hip/hip_runtime.h>

// GCN forward for MI455X (gfx1250, wave32).
// GEMM via V_WMMA_F32_16X16X4_F32 (full fp32 path), edge scatter via
// global_atomic_add_f32, LDS-resident weight matrix, L2-resident graph state.

typedef __attribute__((ext_vector_type(2))) float v2f;
typedef __attribute__((ext_vector_type(8))) float v8f;

#define DIM   128
#define GNUM  64
#define LAYERS 3
#define LNEPS 1e-5f

// ---------------- utility ----------------
__global__ void zero_f32(float* __restrict__ p, int n) {
    int i = blockIdx.x * blockDim.x + threadIdx.x;
    if (i < n) p[i] = 0.0f;
}

// ---------------- degree / norm ----------------
__global__ void deg_accum(const int* __restrict__ ei, float* __restrict__ deg, int E) {
    int e = blockIdx.x * blockDim.x + threadIdx.x;
    if (e < E) atomicAdd(&deg[ei[E + e]], 1.0f);   // dst = ei[E + e]
}

__global__ void deg_to_dinv(float* __restrict__ d, int N) {
    int i = blockIdx.x * blockDim.x + threadIdx.x;
    if (i < N) d[i] = rsqrtf(d[i] + 1.0f);         // +1 self-loop, deg >= 1
}

// ---------------- GEMM: out[N,128] = H[N,128] @ W[128,128] ----------------
// One workgroup = 8 waves; wave w computes rows [strip*16, strip*16+16).
// W is staged in LDS. K-loop in steps of 4 using V_WMMA_F32_16X16X4_F32.
__global__ __launch_bounds__(256) void gemm_wmma(const float* __restrict__ H,
                                                 const float* __restrict__ W,
                                                 float* __restrict__ out, int N) {
    __shared__ float lw[DIM * DIM];                 // 64 KB
    {   // cooperative 128x128 f32 load, float4 vectorized
        const float4* Wv = (const float4*)W;
        float4* lv = (float4*)lw;
        #pragma unroll
        for (int i = 0; i < (DIM * DIM / 4) / 256; ++i)
            lv[threadIdx.x + i * 256] = Wv[threadIdx.x + i * 256];
    }
    __syncthreads();

    const int wave = threadIdx.x >> 5;
    const int lane = threadIdx.x & 31;
    const int half = lane >> 4;                     // 0: K lo pair, 1: K hi pair
    const int ln   = lane & 15;

    const int nstrips = N >> 4;                     // N % 16 == 0 (N = 50000)
    int strip = blockIdx.x * 8 + wave;
    if (strip >= nstrips) strip = nstrips - 1;      // clamp: keep EXEC all-1s for WMMA
    const int rowBase = strip << 4;

    const v2f* arow2 = (const v2f*)(H + (size_t)(rowBase + ln) * DIM);

    v8f acc[8] = {};

    for (int k = 0; k < DIM; k += 4) {
        // A fragment 16x4: lane<16 -> H[row][k..k+1], lane>=16 -> H[row][k+2..k+3]
        v2f a = arow2[(k >> 1) + half];
        const int kb = k + 2 * half;
        #pragma unroll
        for (int t = 0; t < 8; ++t) {
            // B fragment 4x16 of W: lanes stripe columns, VGPRs stripe K
            v2f b;
            b.x = lw[kb * DIM + t * 16 + ln];
            b.y = lw[(kb + 1) * DIM + t * 16 + ln];
            acc[t] = __builtin_amdgcn_wmma_f32_16x16x4_f32(
                false, a, false, b, (short)0, acc[t], false, false);
        }
    }

    // C/D layout: VGPR j -> row (j + 8*half), lane ln -> column
    #pragma unroll
    for (int t = 0; t < 8; ++t) {
        #pragma unroll
        for (int j = 0; j < 8; ++j) {
            int row = rowBase + half * 8 + j;
            out[(size_t)row * DIM + t * 16 + ln] = acc[t][j];
        }
    }
}

// ---------------- acc init: bias + self-loop contribution ----------------
__global__ void acc_init(const float* __restrict__ hw, const float* __restrict__ dinv,
                         const float* __restrict__ bias, float* __restrict__ acc, int N) {
    int i = blockIdx.x * blockDim.x + threadIdx.x;  // over N * 32 float4 groups
    if (i >= N * (DIM / 4)) return;
    int n  = i >> 5;
    int c4 = i & 31;
    float s = dinv[n]; s = s * s;                   // self-loop norm = dinv^2
    float4 hv = ((const float4*)hw)[i];
    float4 bv = ((const float4*)bias)[c4];
    float4 r;
    r.x = bv.x + s * hv.x;  r.y = bv.y + s * hv.y;
    r.z = bv.z + s * hv.z;  r.w = bv.w + s * hv.w;
    ((float4*)acc)[i] = r;
}

// ---------------- edge scatter: wave per edge ----------------
__global__ __launch_bounds__(256) void edge_scatter(const int* __restrict__ ei,
                                                    const float* __restrict__ hw,
                                                    const float* __restrict__ dinv,
                                                    float* __restrict__ acc, int E) {
    int wave = threadIdx.x >> 5, lane = threadIdx.x & 31;
    int e = blockIdx.x * 8 + wave;
    if (e >= E) return;
    int src = ei[e];
    int dst = ei[E + e];
    float w = dinv[src] * dinv[dst];
    float4 v = ((const float4*)(hw + (size_t)src * DIM))[lane];
    float* dp = acc + (size_t)dst * DIM + lane * 4;
    atomicAdd(dp + 0, w * v.x);
    atomicAdd(dp + 1, w * v.y);
    atomicAdd(dp + 2, w * v.z);
    atomicAdd(dp + 3, w * v.w);
}

// ---------------- LayerNorm + ReLU: wave per node ----------------
__global__ __launch_bounds__(256) void ln_relu(const float* __restrict__ acc,
                                               const float* __restrict__ gamma,
                                               const float* __restrict__ beta,
                                               float* __restrict__ out, int N) {
    int wave = threadIdx.x >> 5, lane = threadIdx.x & 31;
    int n = blockIdx.x * 8 + wave;
    if (n >= N) return;
    float4 v = ((const float4*)(acc + (size_t)n * DIM))[lane];
    float s = v.x + v.y + v.z + v.w;
    float q = v.x * v.x + v.y * v.y + v.z * v.z + v.w * v.w;
    #pragma unroll
    for (int off = 16; off > 0; off >>= 1) {
        s += __shfl_xor(s, off, 32);
        q += __shfl_xor(q, off, 32);
    }
    float mu  = s * (1.0f / DIM);
    float var = fmaxf(q * (1.0f / DIM) - mu * mu, 0.0f);
    float inv = rsqrtf(var + LNEPS);
    float4 g = ((const float4*)gamma)[lane];
    float4 b = ((const float4*)beta)[lane];
    float4 r;
    r.x = fmaxf(0.0f, g.x * (v.x - mu) * inv + b.x);
    r.y = fmaxf(0.0f, g.y * (v.y - mu) * inv + b.y);
    r.z = fmaxf(0.0f, g.z * (v.z - mu) * inv + b.z);
    r.w = fmaxf(0.0f, g.w * (v.w - mu) * inv + b.w);
    ((float4*)(out + (size_t)n * DIM))[lane] = r;
}

// ---------------- global mean pool ----------------
__global__ __launch_bounds__(256) void pool_accum(const float* __restrict__ h,
                                                  const int* __restrict__ batch,
                                                  float* __restrict__ pool,
                                                  float* __restrict__ cnt, int N) {
    int wave = threadIdx.x >> 5, lane = threadIdx.x & 31;
    int n = blockIdx.x * 8 + wave;
    if (n >= N) return;
    int g = batch[n];
    float4 v = ((const float4*)(h + (size_t)n * DIM))[lane];
    float* p = pool + (size_t)g * DIM + lane * 4;
    atomicAdd(p + 0, v.x);
    atomicAdd(p + 1, v.y);
    atomicAdd(p + 2, v.z);
    atomicAdd(p + 3, v.w);
    if (lane == 0) atomicAdd(&cnt[g], 1.0f);
}

__global__ void pool_final(const float* __restrict__ pool, const float* __restrict__ cnt,
                           float* __restrict__ out, int total) {
    int i = blockIdx.x * blockDim.x + threadIdx.x;
    if (i < total) out[i] = pool[i] / fmaxf(cnt[i / DIM], 1.0f);
}

// ---------------- host launcher ----------------
static inline size_t al256(size_t x) { return (x + 255) & ~(size_t)255; }

extern "C" void kernel_launch(void* const* d_in, const int* in_sizes, int n_in,
                              void* d_out, int out_size, void* d_ws, size_t ws_size,
                              hipStream_t stream) {
    const float* x      = (const float*)d_in[0];
    const int*   ei     = (const int*)d_in[1];
    const int*   batch  = (const int*)d_in[2];
    const float* Ws     = (const float*)d_in[3];
    const float* bs     = (const float*)d_in[4];
    const float* gammas = (const float*)d_in[5];
    const float* betas  = (const float*)d_in[6];

    const int N = in_sizes[0] / DIM;
    const int E = in_sizes[1] / 2;
    const size_t nd = (size_t)N * DIM * sizeof(float);

    char* ws = (char*)d_ws;
    float* dinv = (float*)ws;                 ws += al256((size_t)N * sizeof(float));
    float* hw   = (float*)ws;                 ws += al256(nd);
    float* accb = (float*)ws;                 ws += al256(nd);
    float* hbuf = (float*)ws;                 ws += al256(nd);
    float* pool = (float*)ws;                 // GNUM*DIM sums + GNUM counts
    float* cnt  = pool + GNUM * DIM;

    // symmetric normalization: dinv = rsqrt(in-degree + 1)
    zero_f32<<<(N + 255) / 256, 256, 0, stream>>>(dinv, N);
    deg_accum<<<(E + 255) / 256, 256, 0, stream>>>(ei, dinv, E);
    deg_to_dinv<<<(N + 255) / 256, 256, 0, stream>>>(dinv, N);

    const int nstrips = N >> 4;
    const float* hcur = x;
    for (int l = 0; l < LAYERS; ++l) {
        gemm_wmma<<<(nstrips + 7) / 8, 256, 0, stream>>>(
            hcur, Ws + (size_t)l * DIM * DIM, hw, N);
        acc_init<<<(N * (DIM / 4) + 255) / 256, 256, 0, stream>>>(
            hw, dinv, bs + (size_t)l * DIM, accb, N);
        edge_scatter<<<(E + 7) / 8, 256, 0, stream>>>(ei, hw, dinv, accb, E);
        ln_relu<<<(N + 7) / 8, 256, 0, stream>>>(
            accb, gammas + (size_t)l * DIM, betas + (size_t)l * DIM, hbuf, N);
        hcur = hbuf;
    }

    zero_f32<<<(GNUM * DIM + GNUM + 255) / 256, 256, 0, stream>>>(pool, GNUM * DIM + GNUM);
    pool_accum<<<(N + 7) / 8, 256, 0, stream>>>(hcur, batch, pool, cnt, N);
    pool_final<<<(GNUM * DIM + 255) / 256, 256, 0, stream>>>(pool, cnt, (float*)d_out, GNUM * DIM);
}